// BiRNNLM_27333171872059
// MI455X (gfx1250) — compile-verified
//
#include <hip/hip_runtime.h>
#include <hip/hip_bf16.h>
#include <math.h>

#define S_LEN 128
#define B_SZ  32
#define EMB   32
#define HID   8
#define VOCAB 32000
#define KDIM  16              // 2*HID
#define WAVES 16              // waves per logits block
#define CHUNKS (VOCAB/16)     // 2000 column chunks
#define CPW    (CHUNKS/WAVES) // 125 chunks per wave

typedef float v2f __attribute__((ext_vector_type(2)));
typedef float v8f __attribute__((ext_vector_type(8)));

// ---------------------------------------------------------------------------
// Kernel 1: bidirectional Elman scans.
// Block 0 = forward scan, block 1 = backward scan. 256 threads: lane (b,j),
// b = t/8, j = t%8. The 8 lanes of one batch element live in one wave, so the
// recurrent h[b][k] exchange is done with __shfl (no barriers, no LDS).
// Embedding rows are fetched directly from the L2-resident lookup table.
// H table layout: H[(s*B + b)*16 + j]      = Hf_table (hidden BEFORE x[s])
//                 H[(s*B + b)*16 + 8 + j]  = Hb_table
// ---------------------------------------------------------------------------
__global__ __launch_bounds__(256)
void birnn_scan_kernel(const int* __restrict__ tok,
                       const float* __restrict__ lookup,
                       const float* __restrict__ Wxf, const float* __restrict__ Whf,
                       const float* __restrict__ Wxb, const float* __restrict__ Whb,
                       const float* __restrict__ Hf0, const float* __restrict__ Hb0,
                       const float* __restrict__ bias_x,
                       const float* __restrict__ bias_hf,
                       const float* __restrict__ bias_hb,
                       float* __restrict__ H)
{
    const int t    = threadIdx.x;        // 0..255
    const int b    = t >> 3;             // batch element
    const int j    = t & 7;              // hidden unit
    const int lane = t & 31;
    const bool bwd = (blockIdx.x != 0);

    const float* Wx = bwd ? Wxb : Wxf;
    const float* Wh = bwd ? Whb : Whf;
    const float* h0 = bwd ? Hb0 : Hf0;
    const float bsum = bias_x[j] + (bwd ? bias_hb[j] : bias_hf[j]);
    const int   off  = bwd ? HID : 0;

    // Per-lane weight columns live in VGPRs.
    float wx[EMB];
#pragma unroll
    for (int e = 0; e < EMB; ++e) wx[e] = Wx[e*HID + j];
    float wh[HID];
#pragma unroll
    for (int k = 0; k < HID; ++k) wh[k] = Wh[k*HID + j];

    float h = h0[j];
    const int gbase = lane & ~7;         // first lane of this b-group

    for (int step = 0; step < S_LEN; ++step) {
        const int s = bwd ? (S_LEN - 1 - step) : step;

        // Table stores hidden state BEFORE consuming x[s] (matches reference shift).
        H[(s*B_SZ + b)*KDIM + off + j] = h;

        const int tk = tok[s*B_SZ + b];
        const float4* xr = (const float4*)(lookup + (size_t)tk * EMB);

        float acc = bsum;
#pragma unroll
        for (int q = 0; q < EMB/4; ++q) {
            float4 xv = xr[q];
            acc = fmaf(xv.x, wx[4*q+0], acc);
            acc = fmaf(xv.y, wx[4*q+1], acc);
            acc = fmaf(xv.z, wx[4*q+2], acc);
            acc = fmaf(xv.w, wx[4*q+3], acc);
        }
#pragma unroll
        for (int k = 0; k < HID; ++k) {
            float hk = __shfl(h, gbase + k, 32);    // h[b][k] from sibling lane
            acc = fmaf(hk, wh[k], acc);
        }
        h = tanhf(acc);
    }
}

// ---------------------------------------------------------------------------
// Kernel 2: logits GEMM + log-softmax, WMMA f32 16x16x4 (K=16 via 4 chained
// WMMAs). One block per 16-row tile of the 4096 (s,b) rows; 16 waves split
// the 2000 column chunks (125 each).
// Pass A: streaming online max/sumexp (per-lane, merged once at the end).
// Pass B: recompute tile, write (logit - max - log(sum)) with NT stores.
// ---------------------------------------------------------------------------
__global__ __launch_bounds__(32*WAVES)
void logits_lsm_kernel(const float* __restrict__ H,
                       const float* __restrict__ Wo,   // [16][VOCAB] row-major
                       const float* __restrict__ bo,   // [VOCAB]
                       float* __restrict__ out)        // [4096][VOCAB]
{
    __shared__ float mS[WAVES][16];
    __shared__ float sS[WAVES][16];
    __shared__ float cS[16];

    const int tid  = threadIdx.x;
    const int wave = tid >> 5;
    const int lane = tid & 31;
    const int ln   = lane & 15;
    const int half = lane >> 4;          // 0: rows r / K+0,1 ; 1: rows r+8 / K+2,3
    const int kofs = half * 2;
    const int rowbase = blockIdx.x * 16;

    // A operand: H[rowbase+ln][kg*4 + kofs + {0,1}]  (ISA 16x4 f32 layout)
    v2f a[4];
#pragma unroll
    for (int kg = 0; kg < 4; ++kg) {
        const float* hp = H + (rowbase + ln)*KDIM + kg*4 + kofs;
        a[kg].x = hp[0];
        a[kg].y = hp[1];
    }

    float m[8], sacc[8];
#pragma unroll
    for (int r = 0; r < 8; ++r) { m[r] = -INFINITY; sacc[r] = 0.0f; }

    const int c0 = wave * CPW;

    // ---------------- Pass A: online max / sum of exp ----------------
    for (int c = 0; c < CPW; ++c) {
        const int n  = (c0 + c)*16 + ln;
        if (c + 1 < CPW)                 // warm GL2/WGP$ for next chunk
            __builtin_prefetch(Wo + (size_t)kofs*VOCAB + n + 16, 0, 1);

        const float bn = bo[n];
        v8f d;
#pragma unroll
        for (int r = 0; r < 8; ++r) d[r] = bn;   // fold bias into C
#pragma unroll
        for (int kg = 0; kg < 4; ++kg) {
            const float* wp = Wo + (size_t)(kg*4 + kofs)*VOCAB + n;
            v2f bv; bv.x = wp[0]; bv.y = wp[VOCAB];
            d = __builtin_amdgcn_wmma_f32_16x16x4_f32(false, a[kg], false, bv,
                                                      (short)0, d, false, false);
        }
#pragma unroll
        for (int r = 0; r < 8; ++r) {    // per-lane online softmax update
            float x  = d[r];
            float nm = fmaxf(m[r], x);
            sacc[r]  = sacc[r]*__expf(m[r]-nm) + __expf(x-nm);
            m[r]     = nm;
        }
    }

    // Merge the 16 lanes of each half (all hold the same row, different cols).
#pragma unroll
    for (int r = 0; r < 8; ++r) {
        float mm = m[r], ss = sacc[r];
#pragma unroll
        for (int off = 1; off < 16; off <<= 1) {
            float om = __shfl_xor(mm, off, 32);
            float os = __shfl_xor(ss, off, 32);
            float nm = fmaxf(mm, om);
            ss = ss*__expf(mm-nm) + os*__expf(om-nm);
            mm = nm;
        }
        m[r] = mm; sacc[r] = ss;
    }
    if (ln == 0) {
#pragma unroll
        for (int r = 0; r < 8; ++r) {
            mS[wave][r + half*8] = m[r];
            sS[wave][r + half*8] = sacc[r];
        }
    }
    __syncthreads();

    // Merge across waves; c[row] = max + log(sumexp).
    if (tid < 16) {
        float mm = mS[0][tid], ss = sS[0][tid];
        for (int w = 1; w < WAVES; ++w) {
            float om = mS[w][tid], os = sS[w][tid];
            float nm = fmaxf(mm, om);
            ss = ss*__expf(mm-nm) + os*__expf(om-nm);
            mm = nm;
        }
        cS[tid] = mm + __logf(ss);
    }
    __syncthreads();

    float cv[8];
#pragma unroll
    for (int r = 0; r < 8; ++r) cv[r] = cS[r + half*8];

    // ---------------- Pass B: recompute + normalized write ----------------
    for (int c = 0; c < CPW; ++c) {
        const int n  = (c0 + c)*16 + ln;
        const float bn = bo[n];
        v8f d;
#pragma unroll
        for (int r = 0; r < 8; ++r) d[r] = bn;
#pragma unroll
        for (int kg = 0; kg < 4; ++kg) {
            const float* wp = Wo + (size_t)(kg*4 + kofs)*VOCAB + n;
            v2f bv; bv.x = wp[0]; bv.y = wp[VOCAB];
            d = __builtin_amdgcn_wmma_f32_16x16x4_f32(false, a[kg], false, bv,
                                                      (short)0, d, false, false);
        }
#pragma unroll
        for (int r = 0; r < 8; ++r) {
            const int row = rowbase + r + half*8;
            __builtin_nontemporal_store(d[r] - cv[r],
                                        out + (size_t)row*VOCAB + n);
        }
    }
}

// ---------------------------------------------------------------------------
extern "C" void kernel_launch(void* const* d_in, const int* in_sizes, int n_in,
                              void* d_out, int out_size, void* d_ws, size_t ws_size,
                              hipStream_t stream)
{
    const int*   tok     = (const int*)  d_in[0];   // input_batch [S,B]
    const float* lookup  = (const float*)d_in[1];   // [V,EMB]
    const float* Wxf     = (const float*)d_in[2];   // [EMB,HID]
    const float* Whf     = (const float*)d_in[3];   // [HID,HID]
    const float* Wxb     = (const float*)d_in[4];
    const float* Whb     = (const float*)d_in[5];
    const float* Wo      = (const float*)d_in[6];   // [2*HID, V]
    const float* Hf0     = (const float*)d_in[7];   // [HID]
    const float* Hb0     = (const float*)d_in[8];
    const float* bias_x  = (const float*)d_in[9];
    const float* bias_hf = (const float*)d_in[10];
    const float* bias_hb = (const float*)d_in[11];
    const float* bias_o  = (const float*)d_in[12];

    float* H   = (float*)d_ws;                      // [S*B][16] = 256 KB
    float* out = (float*)d_out;                     // [S*B][V]

    birnn_scan_kernel<<<2, 256, 0, stream>>>(tok, lookup, Wxf, Whf, Wxb, Whb,
                                             Hf0, Hb0, bias_x, bias_hf, bias_hb, H);

    logits_lsm_kernel<<<(S_LEN*B_SZ)/16, 32*WAVES, 0, stream>>>(H, Wo, bias_o, out);
}